// LinearTransformation_45878840656436
// MI455X (gfx1250) — compile-verified
//
#include <hip/hip_runtime.h>
#include <stdint.h>

// Trilinear affine resample, [B,128,128,128,1] f32, zeros padding.
// Strategy: 8x8x8 output tile per 256-thread block; exact affine AABB of the
// tile in input space; stage AABB into LDS with ONE Tensor-Data-Mover
// (tensor_load_to_lds) 3D-tile DMA; 8-point trilinear gather from LDS.
// Global-gather fallback if box > LDS cap (essentially never for |dT|~0.1).

static constexpr int SD   = 128;   // volume side
static constexpr int TS   = 8;     // output tile side
static constexpr int CAPF = 8192;  // LDS staging capacity in floats (32 KB)

typedef unsigned int u32x4 __attribute__((ext_vector_type(4)));
typedef unsigned int u32x8 __attribute__((ext_vector_type(8)));

__device__ __forceinline__ int   imin(int a, int b)   { return a < b ? a : b; }
__device__ __forceinline__ int   imax(int a, int b)   { return a > b ? a : b; }
__device__ __forceinline__ float rmin(float c, float lo, float hi) { return fminf(c * lo, c * hi); }
__device__ __forceinline__ float rmax(float c, float lo, float hi) { return fmaxf(c * lo, c * hi); }

extern "C" __global__ __launch_bounds__(256)
void trilerp_affine_kernel(const float* __restrict__ img,
                           const float* __restrict__ tf,
                           float* __restrict__ out)
{
    __shared__ float sh[CAPF];

    const int tx = blockIdx.x;
    const int ty = blockIdx.y;
    const int tz = blockIdx.z & 15;
    const int b  = blockIdx.z >> 4;

    // Per-batch affine (wave-uniform -> scalar loads)
    const float* T = tf + b * 12;
    const float t00 = T[0], t01 = T[1], t02 = T[2],  t03 = T[3];
    const float t10 = T[4], t11 = T[5], t12 = T[6],  t13 = T[7];
    const float t20 = T[8], t21 = T[9], t22 = T[10], t23 = T[11];

    const float K = 2.0f / 127.0f;
    const float gxl = fmaf((float)(tx * TS),          K, -1.0f);
    const float gxh = fmaf((float)(tx * TS + TS - 1), K, -1.0f);
    const float gyl = fmaf((float)(ty * TS),          K, -1.0f);
    const float gyh = fmaf((float)(ty * TS + TS - 1), K, -1.0f);
    const float gzl = fmaf((float)(tz * TS),          K, -1.0f);
    const float gzh = fmaf((float)(tz * TS + TS - 1), K, -1.0f);

    // Exact AABB of sample coords over the tile (affine => per-term extremes),
    // padded by 1 voxel against rounding differences vs. the per-voxel fma chain.
    const float ixm = 63.5f * (rmin(t00,gxl,gxh) + rmin(t01,gyl,gyh) + rmin(t02,gzl,gzh) + t03 + 1.0f);
    const float ixM = 63.5f * (rmax(t00,gxl,gxh) + rmax(t01,gyl,gyh) + rmax(t02,gzl,gzh) + t03 + 1.0f);
    const float iym = 63.5f * (rmin(t10,gxl,gxh) + rmin(t11,gyl,gyh) + rmin(t12,gzl,gzh) + t13 + 1.0f);
    const float iyM = 63.5f * (rmax(t10,gxl,gxh) + rmax(t11,gyl,gyh) + rmax(t12,gzl,gzh) + t13 + 1.0f);
    const float izm = 63.5f * (rmin(t20,gxl,gxh) + rmin(t21,gyl,gyh) + rmin(t22,gzl,gzh) + t23 + 1.0f);
    const float izM = 63.5f * (rmax(t20,gxl,gxh) + rmax(t21,gyl,gyh) + rmax(t22,gzl,gzh) + t23 + 1.0f);

    const int blox = imax(0, (int)floorf(ixm) - 1);
    const int bhix = imin(SD - 1, (int)floorf(ixM) + 2);
    const int bloy = imax(0, (int)floorf(iym) - 1);
    const int bhiy = imin(SD - 1, (int)floorf(iyM) + 2);
    const int bloz = imax(0, (int)floorf(izm) - 1);
    const int bhiz = imin(SD - 1, (int)floorf(izM) + 2);
    const int ex = bhix - blox + 1;
    const int ey = bhiy - bloy + 1;
    const int ez = bhiz - bloz + 1;

    const int lx = threadIdx.x & 7;
    const int ly = (threadIdx.x >> 3) & 7;
    const int lz = threadIdx.x >> 6;          // 0..3, each thread does z and z+4
    const int x  = tx * TS + lx;
    const int y  = ty * TS + ly;

    float*       outb = out + ((size_t)b << 21);
    const float* imgb = img + ((size_t)b << 21);

    // Whole tile samples outside the volume -> zeros (zeros padding mode).
    if (ex <= 0 || ey <= 0 || ez <= 0) {
        for (int k = 0; k < 2; ++k) {
            const int z = tz * TS + lz + 4 * k;
            outb[((size_t)z << 14) + (y << 7) + x] = 0.0f;
        }
        return;
    }

    const int  total  = ex * ey * ez;
    const bool useLds = (total <= CAPF);

    if (useLds) {
        // Stage the box with ONE Tensor Data Mover DMA (3D tile of a 3D tensor).
        // TDM ignores EXEC, so gate the issue behind a *scalar* (uniform) branch
        // taken only by wave 0.
        if (__builtin_amdgcn_readfirstlane(threadIdx.x >> 5) == 0) {
            const float* boxp = imgb + (((size_t)bloz << 14) + (size_t)(bloy << 7) + (size_t)blox);
            const unsigned long long ga = (unsigned long long)(uintptr_t)boxp;
            const unsigned lds0 = (unsigned)(uintptr_t)(&sh[0]);

            u32x4 g0;
            g0[0] = 0x1u;                                           // count=1, user desc
            g0[1] = lds0;                                           // lds_addr (bytes)
            g0[2] = (unsigned)ga;                                   // global_addr[31:0]
            g0[3] = ((unsigned)(ga >> 32) & 0x01FFFFFFu) | (2u << 30); // addr[56:32] | type=2

            u32x8 g1;
            g1[0] = 2u << 16;                                       // wg_mask=0, data_size=4B
            g1[1] = 128u << 16;                                     // tensor_dim0 lo16 (bit48)
            g1[2] = 128u << 16;                                     // dim0 hi16=0 | tensor_dim1 lo16
            g1[3] = ((unsigned)ex & 0xFFFFu) << 16;                 // dim1 hi16=0 | tile_dim0
            g1[4] = ((unsigned)ey & 0xFFFFu) | (((unsigned)ez & 0xFFFFu) << 16); // tile_dim1|tile_dim2
            g1[5] = 128u;                                           // tensor_dim0_stride lo32
            g1[6] = 0x4000u << 16;                                  // stride0 hi16=0 | stride1 lo16 (16384)
            g1[7] = 0u;                                             // tensor_dim1_stride hi32

            u32x4 g2;
            g2[0] = 128u;                                           // tensor_dim2
            g2[1] = 0u;                                             // tensor_dim3 (unused)
            g2[2] = 0u;                                             // tensor_dim2_stride lo (unused)
            g2[3] = 0u;                                             // stride2 hi | tile_dim3=0

            u32x4 g3;
            g3[0] = 0u; g3[1] = 0u; g3[2] = 0u; g3[3] = 0u;         // dim3 stride / dim4 / tile_dim4

            asm volatile("tensor_load_to_lds %0, %1, %2, %3"
                         :: "s"(g0), "s"(g1), "s"(g2), "s"(g3) : "memory");
#if __has_builtin(__builtin_amdgcn_s_wait_tensorcnt)
            __builtin_amdgcn_s_wait_tensorcnt(0);
#else
            asm volatile("s_wait_tensorcnt 0x0" ::: "memory");
#endif
        }
        __syncthreads();
    }

    const int row   = ex;
    const int slice = ex * ey;

    for (int k = 0; k < 2; ++k) {
        const int z = tz * TS + lz + 4 * k;

        const float gx = fmaf((float)x, K, -1.0f);
        const float gy = fmaf((float)y, K, -1.0f);
        const float gz = fmaf((float)z, K, -1.0f);

        const float px = fmaf(t02, gz, fmaf(t01, gy, fmaf(t00, gx, t03)));
        const float py = fmaf(t12, gz, fmaf(t11, gy, fmaf(t10, gx, t13)));
        const float pz = fmaf(t22, gz, fmaf(t21, gy, fmaf(t20, gx, t23)));
        const float ix = fmaf(63.5f, px, 63.5f);
        const float iy = fmaf(63.5f, py, 63.5f);
        const float iz = fmaf(63.5f, pz, 63.5f);

        const float valid = (ix >= 0.0f && ix <= 127.0f &&
                             iy >= 0.0f && iy <= 127.0f &&
                             iz >= 0.0f && iz <= 127.0f) ? 1.0f : 0.0f;

        const float fx = fminf(fmaxf(floorf(ix), 0.0f), 127.0f);
        const float fy = fminf(fmaxf(floorf(iy), 0.0f), 127.0f);
        const float fz = fminf(fmaxf(floorf(iz), 0.0f), 127.0f);
        const int ix0 = (int)fx, iy0 = (int)fy, iz0 = (int)fz;

        // Opposite-corner abs weights (|a*b| == |a|*|b| exactly in IEEE).
        const float dx = ix - fx, dy = iy - fy, dz = iz - fz;
        const float ax1 = fabsf(dx), ax0 = fabsf(dx - 1.0f);
        const float ay1 = fabsf(dy), ay0 = fabsf(dy - 1.0f);
        const float az1 = fabsf(dz), az0 = fabsf(dz - 1.0f);

        float r;
        if (useLds) {
            const int rx0 = imin(imax(ix0     - blox, 0), ex - 1);
            const int rx1 = imin(imax(ix0 + 1 - blox, 0), ex - 1);
            const int ry0 = imin(imax(iy0     - bloy, 0), ey - 1);
            const int ry1 = imin(imax(iy0 + 1 - bloy, 0), ey - 1);
            const int rz0 = imin(imax(iz0     - bloz, 0), ez - 1);
            const int rz1 = imin(imax(iz0 + 1 - bloz, 0), ez - 1);
            const int p00 = rz0 * slice + ry0 * row;
            const int p01 = rz0 * slice + ry1 * row;
            const int p10 = rz1 * slice + ry0 * row;
            const int p11 = rz1 * slice + ry1 * row;
            r = ax0*ay0*az0 * sh[p00 + rx0] + ax1*ay0*az0 * sh[p00 + rx1]
              + ax0*ay1*az0 * sh[p01 + rx0] + ax1*ay1*az0 * sh[p01 + rx1]
              + ax0*ay0*az1 * sh[p10 + rx0] + ax1*ay0*az1 * sh[p10 + rx1]
              + ax0*ay1*az1 * sh[p11 + rx0] + ax1*ay1*az1 * sh[p11 + rx1];
        } else {
            const int x1 = imin(ix0 + 1, 127);
            const int y1 = imin(iy0 + 1, 127);
            const int z1 = imin(iz0 + 1, 127);
            const int q00 = (iz0 << 14) + (iy0 << 7);
            const int q01 = (iz0 << 14) + (y1  << 7);
            const int q10 = (z1  << 14) + (iy0 << 7);
            const int q11 = (z1  << 14) + (y1  << 7);
            r = ax0*ay0*az0 * imgb[q00 + ix0] + ax1*ay0*az0 * imgb[q00 + x1]
              + ax0*ay1*az0 * imgb[q01 + ix0] + ax1*ay1*az0 * imgb[q01 + x1]
              + ax0*ay0*az1 * imgb[q10 + ix0] + ax1*ay0*az1 * imgb[q10 + x1]
              + ax0*ay1*az1 * imgb[q11 + ix0] + ax1*ay1*az1 * imgb[q11 + x1];
        }

        outb[((size_t)z << 14) + (y << 7) + x] = r * valid;
    }
}

extern "C" void kernel_launch(void* const* d_in, const int* in_sizes, int n_in,
                              void* d_out, int out_size, void* d_ws, size_t ws_size,
                              hipStream_t stream) {
    const float* img = (const float*)d_in[0];
    const float* tf  = (const float*)d_in[1];
    float* out = (float*)d_out;
    const int B = in_sizes[1] / 12;                       // transfos: [B,3,4]
    dim3 grid(SD / TS, SD / TS, (SD / TS) * B);           // 16 x 16 x 16*B tiles
    trilerp_affine_kernel<<<grid, 256, 0, stream>>>(img, tf, out);
}